// DeepPSDual_45878840656229
// MI455X (gfx1250) — compile-verified
//
#include <hip/hip_runtime.h>
#include <math.h>
#include <stdint.h>

typedef __attribute__((ext_vector_type(2))) float v2f;
typedef __attribute__((ext_vector_type(8))) float v8f;

#define B_    256
#define TR    512
#define PDIM  128
#define HDIM  128
#define R_    (B_*TR)
#define N3H   384
#define K2P   256
#define TA    64
#define PSTD  32
#define PSTAT 16
#define FEAT  176
#define HH    64
#define HSTR  132   // padded LDS row stride (floats): stride%64==4 -> conflict-free 16-lane access

// LDS layout (floats) for scan kernel
#define LWHH  (N3H * HSTR)          // 50688
#define LH    (16 * HSTR)           // 2112
#define LGH   (16 * N3H)            // 6144
#define LBHH  (N3H)                 // 384
#define LDH   (2 * 16 * PDIM)       // 4096  : [delta(16x128) | htil(16x128)] single step buffer
#define LGI   (2 * 16 * N3H)        // 12288 : gi double buffer, 16x384 each
#define LTOT  (LWHH + LH + LGH + LBHH + LDH + LGI)   // 75712 floats = 302848 B <= 320KB WGP LDS

__device__ __forceinline__ float sigmoidf_(float x) { return 1.0f / (1.0f + __expf(-x)); }

// Async global->LDS copy, 16B per lane, GVS addressing (uniform SGPR base +
// per-lane signed 32-bit byte offset). Tracked by ASYNCcnt.
__device__ __forceinline__ void async_b128(uint32_t lds_byte_addr, uint32_t voff_bytes,
                                           const float* sbase) {
    asm volatile("global_load_async_to_lds_b128 %0, %1, %2 offset:0"
                 :: "v"(lds_byte_addr), "v"(voff_bytes), "s"(sbase)
                 : "memory");
}
__device__ __forceinline__ void wait_async0() {
    asm volatile("s_wait_asynccnt 0x0" ::: "memory");
}
__device__ __forceinline__ uint32_t lds_u32(const void* p) {
    return (uint32_t)(uintptr_t)p;   // LDS aperture: addr[31:0] == LDS byte offset
}

// ---------------------------------------------------------------------------
// Kernel A: elementwise precompute.
// delta = exp(-softplus(dt)) = sigmoid(-dt); x_hat; x_dec; inp = [x_dec | m]
// ---------------------------------------------------------------------------
__global__ void prep_kernel(const float* __restrict__ X, const float* __restrict__ M,
                            const float* __restrict__ DT, const float* __restrict__ xmean,
                            float* __restrict__ delta, float* __restrict__ xhat,
                            float* __restrict__ inp) {
    size_t e = (size_t)blockIdx.x * blockDim.x + threadIdx.x;
    size_t total = (size_t)R_ * PDIM;
    if (e >= total) return;
    int p = (int)(e & (PDIM - 1));
    size_t r = e >> 7;
    float x = X[e], m = M[e], dt = DT[e];
    float d  = 1.0f / (1.0f + __expf(dt));          // sigmoid(-dt)
    float xm = xmean[p];
    float xh = m * x + (1.0f - m) * xm;
    float xd = m * x + (1.0f - m) * (d * xh + (1.0f - d) * xm);
    delta[e] = d;
    xhat[e]  = xh;
    inp[r * K2P + p]        = xd;
    inp[r * K2P + PDIM + p] = m;
}

// ---------------------------------------------------------------------------
// Generic fp32 WMMA GEMM:  Out[r][n] = act( sum_k A[r][k]*W[n][k] + bias[n] )
// One wave32 per block computes a 16x64 tile using V_WMMA_F32_16X16X4_F32.
// ---------------------------------------------------------------------------
template <int KDIM, int ACT>
__global__ void wmma_gemm_bias_act(const float* __restrict__ A, const float* __restrict__ W,
                                   const float* __restrict__ bias, float* __restrict__ Out,
                                   int N) {
    int lane  = threadIdx.x & 31;
    int rowA  = lane & 15;
    int khalf = lane >> 4;
    int lcol  = lane & 15;
    int lhalf = lane >> 4;
    size_t r0 = (size_t)blockIdx.x * 16;
    int n0 = blockIdx.y * 64;
    const float* arow = A + (r0 + rowA) * KDIM;
    v8f acc[4] = {v8f{}, v8f{}, v8f{}, v8f{}};
#pragma unroll 4
    for (int k = 0; k < KDIM; k += 4) {
        int ks = k + 2 * khalf;
        v2f a = {arow[ks], arow[ks + 1]};
#pragma unroll
        for (int t = 0; t < 4; ++t) {
            const float* wrow = W + (size_t)(n0 + t * 16 + lcol) * KDIM + ks;
            v2f b = {wrow[0], wrow[1]};
            acc[t] = __builtin_amdgcn_wmma_f32_16x16x4_f32(
                false, a, false, b, (short)0, acc[t], false, false);
        }
    }
#pragma unroll
    for (int t = 0; t < 4; ++t) {
        int ncol = n0 + t * 16 + lcol;
        float bv = bias[ncol];
#pragma unroll
        for (int i = 0; i < 8; ++i) {
            float v = acc[t][i] + bv;
            if (ACT == 1) v = tanhf(v);
            Out[(r0 + i + 8 * lhalf) * (size_t)N + ncol] = v;
        }
    }
}

// ---------------------------------------------------------------------------
// Kernel D: sequential GRU scan. 16 blocks x 256 threads (8 wave32s).
// Block owns 16 batch rows for all 512 steps. Whh resident in LDS;
// per-step streams (delta/htil/gi) prefetched with async global->LDS b128
// copies (gi double-buffered, delta/htil single-buffered), hidden behind the WMMA phase.
// ---------------------------------------------------------------------------
__global__ void scan_kernel(const float* __restrict__ delta, const float* __restrict__ htil,
                            const float* __restrict__ gi, const float* __restrict__ Whh,
                            const float* __restrict__ bhh, float* __restrict__ Hraw) {
    extern __shared__ __align__(16) float smem[];
    float* whhL = smem;                  // LWHH
    float* hL   = whhL + LWHH;           // LH
    float* ghL  = hL + LH;               // LGH
    float* bhhL = ghL + LGH;             // LBHH
    float* dhL  = bhhL + LBHH;           // LDH  : delta | htil for current step
    float* giL  = dhL + LDH;             // LGI  : gi double buffer

    int tid   = threadIdx.x;
    int bbase = blockIdx.x * 16;

    const uint32_t dhA = lds_u32(dhL);
    const uint32_t giA = lds_u32(giL);

    // ---- prologue: kick off async prefetch of step 0 streams, then stage Whh
    {
        const float* gbase = gi + ((size_t)bbase * TR + 0) * N3H;
        for (int i = tid; i < 1536; i += 256) {           // 16 rows x 384 f = 1536 x b128
            int row = i / 96, c = (i % 96) << 2;
            async_b128(giA + (uint32_t)((row * N3H + c) << 2),
                       (uint32_t)((row * (TR * N3H) + c) << 2), gbase);
        }
        const float* dbase = delta + ((size_t)bbase * TR + 0) * PDIM;
        const float* hbase = htil  + ((size_t)bbase * TR + 0) * PDIM;
        for (int i = tid; i < 512; i += 256) {            // 16 rows x 128 f = 512 x b128
            int row = i >> 5, c = (i & 31) << 2;
            uint32_t vo = (uint32_t)((row * (TR * PDIM) + c) << 2);
            async_b128(dhA + (uint32_t)((row * PDIM + c) << 2), vo, dbase);
            async_b128(dhA + (uint32_t)(((16 * PDIM) + row * PDIM + c) << 2), vo, hbase);
        }
    }
    for (int idx = tid; idx < N3H * HDIM; idx += 256) {
        int n = idx >> 7, k = idx & 127;
        whhL[n * HSTR + k] = Whh[idx];
    }
    for (int idx = tid; idx < N3H; idx += 256) bhhL[idx] = bhh[idx];
    for (int idx = tid; idx < 16 * HSTR; idx += 256) hL[idx] = 0.0f;
    wait_async0();
    __syncthreads();

    int lane  = tid & 31;
    int wave  = tid >> 5;
    int rowA  = lane & 15;
    int khalf = lane >> 4;
    int lcol  = lane & 15;
    int lhalf = lane >> 4;

    for (int t = 0; t < TR; ++t) {
        int buf = t & 1;

        // prefetch gi for t+1 into the alternate buffer (overlaps whole step)
        if (t + 1 < TR) {
            const float* gbase = gi + ((size_t)bbase * TR + (t + 1)) * N3H;
            uint32_t dstA = giA + (uint32_t)(((buf ^ 1) * 16 * N3H) << 2);
            for (int i = tid; i < 1536; i += 256) {
                int row = i / 96, c = (i % 96) << 2;
                async_b128(dstA + (uint32_t)((row * N3H + c) << 2),
                           (uint32_t)((row * (TR * N3H) + c) << 2), gbase);
            }
        }

        // phase 1: decay toward h_til: h = d*h + (1-d)*h_til   (from LDS stage)
        for (int e = tid; e < 16 * HDIM; e += 256) {
            int row = e >> 7, p = e & 127;
            float d = dhL[row * PDIM + p];
            hL[row * HSTR + p] = d * hL[row * HSTR + p]
                               + (1.0f - d) * dhL[16 * PDIM + row * PDIM + p];
        }
        __syncthreads();

        // dh buffer is free now: prefetch delta/htil for t+1 (overlaps WMMA + gates)
        if (t + 1 < TR) {
            const float* dbase = delta + ((size_t)bbase * TR + (t + 1)) * PDIM;
            const float* hbase = htil  + ((size_t)bbase * TR + (t + 1)) * PDIM;
            for (int i = tid; i < 512; i += 256) {
                int row = i >> 5, c = (i & 31) << 2;
                uint32_t vo = (uint32_t)((row * (TR * PDIM) + c) << 2);
                async_b128(dhA + (uint32_t)((row * PDIM + c) << 2), vo, dbase);
                async_b128(dhA + (uint32_t)(((16 * PDIM) + row * PDIM + c) << 2), vo, hbase);
            }
        }

        // phase 2: gh = h @ Whh^T + bhh via fp32 WMMA; 3 of 24 tiles per wave
#pragma unroll
        for (int tt = 0; tt < 3; ++tt) {
            int tile = wave * 3 + tt;
            int ncol = tile * 16 + lcol;
            v8f acc = {};
#pragma unroll
            for (int k = 0; k < HDIM; k += 4) {
                int ks = k + 2 * khalf;
                v2f a = {hL[rowA * HSTR + ks], hL[rowA * HSTR + ks + 1]};
                v2f b = {whhL[ncol * HSTR + ks], whhL[ncol * HSTR + ks + 1]};
                acc = __builtin_amdgcn_wmma_f32_16x16x4_f32(
                    false, a, false, b, (short)0, acc, false, false);
            }
            float bv = bhhL[ncol];
#pragma unroll
            for (int i = 0; i < 8; ++i)
                ghL[(i + 8 * lhalf) * N3H + ncol] = acc[i] + bv;
        }
        __syncthreads();

        // phase 3: GRU gate update from LDS-staged gi; write new h + H_raw
        const float* giC = giL + buf * 16 * N3H;
        for (int e = tid; e < 16 * HDIM; e += 256) {
            int row = e >> 7, j = e & 127;
            size_t g = ((size_t)(bbase + row)) * TR + t;
            float rr = sigmoidf_(giC[row * N3H + j]       + ghL[row * N3H + j]);
            float zz = sigmoidf_(giC[row * N3H + 128 + j] + ghL[row * N3H + 128 + j]);
            float nn = tanhf(giC[row * N3H + 256 + j] + rr * ghL[row * N3H + 256 + j]);
            float hp = hL[row * HSTR + j];
            float hn = (1.0f - zz) * nn + zz * hp;
            hL[row * HSTR + j] = hn;
            Hraw[g * (size_t)PDIM + j] = hn;
        }
        // step boundary: own async ops done, then make them visible to all waves
        wait_async0();
        __syncthreads();
    }
}

// ---------------------------------------------------------------------------
// Kernel E: gather + head MLP (tiny: 184 MFLOP, plain VALU).
// ---------------------------------------------------------------------------
__global__ void head_kernel(const float* __restrict__ Hraw, const int* __restrict__ idx_map,
                            const float* __restrict__ STD, const float* __restrict__ Z,
                            const float* __restrict__ W1, const float* __restrict__ b1,
                            const float* __restrict__ W2,
                            float* __restrict__ eta, float* __restrict__ Hagg,
                            float* __restrict__ maskO) {
    __shared__ float feats[FEAT];
    __shared__ float hid[HH];
    int blk = blockIdx.x;
    int b   = blk >> 6;
    int tid = threadIdx.x;
    int idx = idx_map[blk];
    bool valid = idx >= 0;
    int ic = idx > 0 ? idx : 0;
    const float* hrow = Hraw + (((size_t)b) * TR + ic) * HDIM;
    for (int j = tid; j < HDIM; j += HH) {
        float v = valid ? hrow[j] : 0.0f;
        feats[j] = v;
        Hagg[(size_t)blk * HDIM + j] = v;
    }
    if (tid < PSTD)  feats[HDIM + tid]        = STD[(size_t)blk * PSTD + tid];
    if (tid < PSTAT) feats[HDIM + PSTD + tid] = Z[(size_t)b * PSTAT + tid];
    __syncthreads();
    const float* w1r = W1 + tid * FEAT;
    float acc = b1[tid];
#pragma unroll 8
    for (int k = 0; k < FEAT; ++k) acc += feats[k] * w1r[k];
    hid[tid] = fmaxf(acc, 0.0f) * W2[tid];
    __syncthreads();
    if (tid == 0) {
        float s = 0.0f;
        for (int j = 0; j < HH; ++j) s += hid[j];
        eta[blk]   = s;
        maskO[blk] = valid ? 1.0f : 0.0f;
    }
}

// ---------------------------------------------------------------------------
extern "C" void kernel_launch(void* const* d_in, const int* in_sizes, int n_in,
                              void* d_out, int out_size, void* d_ws, size_t ws_size,
                              hipStream_t stream) {
    const float* X   = (const float*)d_in[0];
    const float* M   = (const float*)d_in[1];
    const float* DT  = (const float*)d_in[2];
    const float* STD = (const float*)d_in[3];
    const float* Z   = (const float*)d_in[4];
    const int*   idx = (const int*)d_in[5];
    const float* xm  = (const float*)d_in[6];
    const float* Wd  = (const float*)d_in[7];
    const float* bd  = (const float*)d_in[8];
    const float* Wih = (const float*)d_in[9];
    const float* bih = (const float*)d_in[10];
    const float* Whh = (const float*)d_in[11];
    const float* bhh = (const float*)d_in[12];
    const float* W1  = (const float*)d_in[13];
    const float* b1  = (const float*)d_in[14];
    const float* W2  = (const float*)d_in[15];

    float* out   = (float*)d_out;
    float* eta   = out;                                  // [B,64]
    float* Hraw  = out  + (size_t)B_ * TA;               // [B,Tr,H]
    float* Hagg  = Hraw + (size_t)B_ * TR * HDIM;        // [B,64,H]
    float* maskO = Hagg + (size_t)B_ * TA * HDIM;        // [B,64]

    float* ws    = (float*)d_ws;
    float* delta = ws;                                   // R*128
    float* xhat  = delta + (size_t)R_ * PDIM;            // R*128
    float* inp   = xhat  + (size_t)R_ * PDIM;            // R*256
    float* htil  = inp   + (size_t)R_ * K2P;             // R*128
    float* gi    = htil  + (size_t)R_ * HDIM;            // R*384

    {   // elementwise precompute
        size_t total = (size_t)R_ * PDIM;
        int thr = 256;
        int blocks = (int)((total + thr - 1) / thr);
        prep_kernel<<<blocks, thr, 0, stream>>>(X, M, DT, xm, delta, xhat, inp);
    }
    {   // Htil = tanh(Xhat @ Wd^T + bd)   [131072 x 128], K=128
        dim3 g(R_ / 16, HDIM / 64);
        wmma_gemm_bias_act<PDIM, 1><<<g, 32, 0, stream>>>(xhat, Wd, bd, htil, HDIM);
    }
    {   // Gi = [Xdec|M] @ Wih^T + bih     [131072 x 384], K=256
        dim3 g(R_ / 16, N3H / 64);
        wmma_gemm_bias_act<K2P, 0><<<g, 32, 0, stream>>>(inp, Wih, bih, gi, N3H);
    }
    {   // sequential GRU scan, Whh + streamed operands resident in LDS
        size_t shmem = (size_t)LTOT * sizeof(float);     // 302,848 B <= 320KB
        scan_kernel<<<B_ / 16, 256, shmem, stream>>>(delta, htil, gi, Whh, bhh, Hraw);
    }
    {   // gather + head MLP
        head_kernel<<<B_ * TA, HH, 0, stream>>>(Hraw, idx, STD, Z, W1, b1, W2,
                                                eta, Hagg, maskO);
    }
}